// my_attention_49503793053990
// MI455X (gfx1250) — compile-verified
//
#include <hip/hip_runtime.h>

// Problem constants (match reference)
#define BB 32
#define TT 4096
#define DD 512
#define KK 128
#define TC 32          // T-chunks for mean pass (128 t per block)
#define NC 32          // T-chunks for flash pass (128 t per block)

typedef float v2f __attribute__((ext_vector_type(2)));
typedef float v4f __attribute__((ext_vector_type(4)));
typedef float v8f __attribute__((ext_vector_type(8)));

// ---- gfx1250 async global->LDS copy (ASYNCcnt-tracked), with fallback ----
#if __has_builtin(__builtin_amdgcn_global_load_async_to_lds_b128)
#define HAVE_ASYNC_LDS 1
typedef int v4i_vs __attribute__((vector_size(16)));   // GCC-style, per builtin sig
#define ASG __attribute__((address_space(1)))
#define ASL __attribute__((address_space(3)))
__device__ __forceinline__ void async_ld_b128(const v4f* gsrc, v4f* ldst) {
    __builtin_amdgcn_global_load_async_to_lds_b128(
        (ASG v4i_vs*)(uintptr_t)gsrc,
        (ASL v4i_vs*)(uint32_t)(uintptr_t)ldst,
        0, 0);
}
#else
#define HAVE_ASYNC_LDS 0
#endif

#if __has_builtin(__builtin_amdgcn_s_wait_asynccnt)
#define WAIT_ASYNC(n) __builtin_amdgcn_s_wait_asynccnt(n)
#else
#define WAIT_ASYNC(n) asm volatile("s_wait_asynccnt %0" ::"i"(n))
#endif

// ---------------------------------------------------------------------------
// K1: partial sums over T.  grid = B*TC blocks x 128 threads.
// ---------------------------------------------------------------------------
__global__ __launch_bounds__(128) void mean_part_kernel(const float* __restrict__ x,
                                                        float* __restrict__ part) {
    const int b = blockIdx.x >> 5;
    const int c = blockIdx.x & 31;
    const int tid = threadIdx.x;
    const v4f* xp = (const v4f*)(x + ((size_t)b * TT + (size_t)c * 128) * DD);
    v4f acc = {0.f, 0.f, 0.f, 0.f};
    for (int t = 0; t < 128; ++t)
        acc += xp[t * (DD / 4) + tid];
    ((v4f*)(part + ((size_t)(b * TC + c)) * DD))[tid] = acc;
}

// ---------------------------------------------------------------------------
// K2: reduce chunks -> mean.  grid = B blocks x 128 threads.
// ---------------------------------------------------------------------------
__global__ __launch_bounds__(128) void mean_final_kernel(const float* __restrict__ part,
                                                         float* __restrict__ s_mean) {
    const int b = blockIdx.x;
    const int tid = threadIdx.x;
    v4f acc = {0.f, 0.f, 0.f, 0.f};
    for (int c = 0; c < TC; ++c)
        acc += ((const v4f*)(part + ((size_t)(b * TC + c)) * DD))[tid];
    ((v4f*)(s_mean + (size_t)b * DD))[tid] = acc * (1.0f / (float)TT);
}

// ---------------------------------------------------------------------------
// K3: fused small-GEMM kernel, single block of 512 threads (16 waves).
//   logits(32x128) = s_mean(32x512) @ kern(512x128)     [WMMA f32 16x16x4]
//   w = softmax_rows(logits)                             [in LDS]
//   C1(32x512)    = w(32x128) @ kern^T(128x512)          [WMMA f32 16x16x4]
// Fragment layouts per CDNA5 ISA 7.12.2:
//   A: lane L holds row m=L&15; VGPR0/1 = K = k0+2*(L>>4) (+1)
//   B: lane L holds col n=L&15; VGPR0/1 = K = k0+2*(L>>4) (+1)
//   C/D: lane L, reg r -> (m = 8*(L>>4)+r, n = L&15)
// ---------------------------------------------------------------------------
__global__ __launch_bounds__(512) void small_gemm_kernel(const float* __restrict__ s_mean,
                                                         const float* __restrict__ kern,
                                                         float* __restrict__ C1) {
    __shared__ float lw[BB * KK];  // logits, then softmax weights (16 KB)
    const int tid   = threadIdx.x;
    const int wv    = tid >> 5;
    const int lane  = tid & 31;
    const int lhalf = lane >> 4;
    const int l15   = lane & 15;

    // ---- GEMM1: logits = s_mean @ kern  (2x8 tiles of 16x16, K=512) ----
    {
        const int mi = wv >> 3;        // 0..1
        const int ni = wv & 7;         // 0..7
        const int m = mi * 16 + l15;
        const int n = ni * 16 + l15;
        v8f acc = {};
        for (int k0 = 0; k0 < DD; k0 += 4) {
            const int kk = k0 + lhalf * 2;
            v2f a, bf;
            a.x  = s_mean[m * DD + kk];
            a.y  = s_mean[m * DD + kk + 1];
            bf.x = kern[kk * KK + n];
            bf.y = kern[(kk + 1) * KK + n];
            acc = __builtin_amdgcn_wmma_f32_16x16x4_f32(false, a, false, bf,
                                                        (short)0, acc, false, false);
        }
        const int mrow = mi * 16 + lhalf * 8;
#pragma unroll
        for (int r = 0; r < 8; ++r)
            lw[(mrow + r) * KK + ni * 16 + l15] = acc[r];
    }
    __syncthreads();

    // ---- softmax over each of 32 rows (128 cols), 16 threads per row ----
    {
        const int row = tid >> 4;
        const int c0  = tid & 15;
        float v[8];
        float mx = -3.4e38f;
#pragma unroll
        for (int j = 0; j < 8; ++j) {
            v[j] = lw[row * KK + c0 + j * 16];
            mx = fmaxf(mx, v[j]);
        }
#pragma unroll
        for (int off = 8; off >= 1; off >>= 1)
            mx = fmaxf(mx, __shfl_xor(mx, off, 32));
        float sum = 0.f;
#pragma unroll
        for (int j = 0; j < 8; ++j) { v[j] = __expf(v[j] - mx); sum += v[j]; }
#pragma unroll
        for (int off = 8; off >= 1; off >>= 1)
            sum += __shfl_xor(sum, off, 32);
        const float inv = 1.0f / sum;
#pragma unroll
        for (int j = 0; j < 8; ++j)
            lw[row * KK + c0 + j * 16] = v[j] * inv;
    }
    __syncthreads();

    // ---- GEMM2: C1 = w @ kern^T  (2x32 tiles, K=128; 4 tiles per wave) ----
#pragma unroll
    for (int i = 0; i < 4; ++i) {
        const int tile = wv + 16 * i;   // 0..63
        const int mi = tile & 1;        // 0..1
        const int ni = tile >> 1;       // 0..31
        const int m = mi * 16 + l15;
        const int n = ni * 16 + l15;    // n indexes d; B(k,n) = kern[n*KK + k]
        v8f acc = {};
        for (int k0 = 0; k0 < KK; k0 += 4) {
            const int kk = k0 + lhalf * 2;
            v2f a, bf;
            a.x  = lw[m * KK + kk];
            a.y  = lw[m * KK + kk + 1];
            bf.x = kern[n * KK + kk];
            bf.y = kern[n * KK + kk + 1];
            acc = __builtin_amdgcn_wmma_f32_16x16x4_f32(false, a, false, bf,
                                                        (short)0, acc, false, false);
        }
        const int mrow = mi * 16 + lhalf * 8;
#pragma unroll
        for (int r = 0; r < 8; ++r)
            C1[(mrow + r) * DD + ni * 16 + l15] = acc[r];
    }
}

// ---------------------------------------------------------------------------
// K4: fused scores + online softmax + weighted accumulation (flash style).
// grid = B*NC blocks x 128 threads (4 waves); each block owns 128 t's.
// Double-buffered: while computing on one 16x512 LDS tile, the next tile is
// in flight via GLOBAL_LOAD_ASYNC_TO_LDS_B128 (ASYNCcnt), retired with
// s_wait_asynccnt 16 (async loads complete in order).  x is read from HBM
// exactly once in this pass.
// ---------------------------------------------------------------------------
__global__ __launch_bounds__(128) void flash_kernel(const float* __restrict__ x,
                                                    const float* __restrict__ C1,
                                                    float* __restrict__ acc_part,
                                                    float* __restrict__ ml_part) {
    __shared__ v4f xs4[2][16 * (DD / 4)];  // 2 x 32 KB
    __shared__ float ssc[16];
    const int tid  = threadIdx.x;
    const int wv   = tid >> 5;
    const int lane = tid & 31;
    const int b = blockIdx.x >> 5;
    const int c = blockIdx.x & 31;
    const int t0 = c * 128;

    const v4f* c1v4 = (const v4f*)(C1 + (size_t)b * DD);
    v4f c1d[4];
#pragma unroll
    for (int j = 0; j < 4; ++j) c1d[j] = c1v4[lane + 32 * j];  // slice for dots

    const v4f* xbase = (const v4f*)(x + ((size_t)b * TT + (size_t)t0) * DD);

#if HAVE_ASYNC_LDS
    // prologue: tile 0 -> buffer 0
#pragma unroll
    for (int t = 0; t < 16; ++t)
        async_ld_b128(xbase + t * (DD / 4) + tid, &xs4[0][t * (DD / 4) + tid]);
#endif

    float m = -3.4e38f, l = 0.f;
    v4f acc = {0.f, 0.f, 0.f, 0.f};

    for (int it = 0; it < 8; ++it) {
        const int cur = it & 1;
#if HAVE_ASYNC_LDS
        if (it < 7) {   // issue next tile into the other buffer
            const v4f* src = xbase + (size_t)(it + 1) * 16 * (DD / 4);
#pragma unroll
            for (int t = 0; t < 16; ++t)
                async_ld_b128(src + t * (DD / 4) + tid,
                              &xs4[cur ^ 1][t * (DD / 4) + tid]);
            WAIT_ASYNC(16);   // oldest 16 (current tile) retired, in order
        } else {
            WAIT_ASYNC(0);
        }
        __syncthreads();
#else
        __syncthreads();
#pragma unroll
        for (int t = 0; t < 16; ++t) {
            const v4f* src = xbase + (size_t)(it * 16 + t) * (DD / 4);
            xs4[cur][t * (DD / 4) + tid] = src[tid];
        }
        __syncthreads();
#endif
        // wave wv computes dots for t = wv*4 .. wv*4+3 from current buffer
#pragma unroll
        for (int k = 0; k < 4; ++k) {
            const int t = wv * 4 + k;
            const v4f* row = &xs4[cur][t * (DD / 4)];
            float d0 = 0.f;
#pragma unroll
            for (int j = 0; j < 4; ++j) {
                const v4f xv = row[lane + 32 * j];
                d0 += xv.x * c1d[j].x + xv.y * c1d[j].y +
                      xv.z * c1d[j].z + xv.w * c1d[j].w;
            }
#pragma unroll
            for (int off = 16; off >= 1; off >>= 1)
                d0 += __shfl_xor(d0, off, 32);
            if (lane == 0) ssc[t] = d0;
        }
        __syncthreads();
        // online-softmax update (every thread computes identical m/l)
        float sc[16];
        float tmax = -3.4e38f;
#pragma unroll
        for (int t = 0; t < 16; ++t) { sc[t] = ssc[t]; tmax = fmaxf(tmax, sc[t]); }
        const float mn = fmaxf(m, tmax);
        const float f = __expf(m - mn);   // first iter: exp(-inf - finite) = 0
        l *= f;
        acc *= f;
#pragma unroll
        for (int t = 0; t < 16; ++t) {
            const float p = __expf(sc[t] - mn);
            l += p;
            acc += p * xs4[cur][t * (DD / 4) + tid];
        }
        m = mn;
        __syncthreads();   // all reads of 'cur' done before it is overwritten
    }

    ((v4f*)(acc_part + ((size_t)(b * NC + c)) * DD))[tid] = acc;
    if (tid == 0) {
        ml_part[(b * NC + c) * 2 + 0] = m;
        ml_part[(b * NC + c) * 2 + 1] = l;
    }
}

// ---------------------------------------------------------------------------
// K5: merge flash partials across chunks.  grid = B x 128 threads.
// ---------------------------------------------------------------------------
__global__ __launch_bounds__(128) void merge_kernel(const float* __restrict__ acc_part,
                                                    const float* __restrict__ ml_part,
                                                    float* __restrict__ out) {
    const int b = blockIdx.x;
    const int tid = threadIdx.x;
    float M = -3.4e38f;
    for (int c = 0; c < NC; ++c)
        M = fmaxf(M, ml_part[(b * NC + c) * 2]);
    float L = 0.f;
    v4f acc = {0.f, 0.f, 0.f, 0.f};
    for (int c = 0; c < NC; ++c) {
        const float w = __expf(ml_part[(b * NC + c) * 2] - M);
        L += w * ml_part[(b * NC + c) * 2 + 1];
        acc += w * ((const v4f*)(acc_part + ((size_t)(b * NC + c)) * DD))[tid];
    }
    const float inv = 1.0f / L;
    ((v4f*)(out + (size_t)b * DD))[tid] = acc * inv;
}

// ---------------------------------------------------------------------------
// Workspace layout (floats):
//   [0          .. 524287]  part / acc_part  (B*32*512)  -- reused
//   [524288     .. 540671]  s_mean           (B*D)
//   [540672     .. 557055]  C1               (B*D)
//   [557056     .. 559103]  ml_part          (B*32*2)
// total ~2.24 MB
// ---------------------------------------------------------------------------
extern "C" void kernel_launch(void* const* d_in, const int* in_sizes, int n_in,
                              void* d_out, int out_size, void* d_ws, size_t ws_size,
                              hipStream_t stream) {
    const float* x    = (const float*)d_in[0];   // (B,T,D) fp32
    const float* kern = (const float*)d_in[1];   // (D,K)   fp32
    float* out = (float*)d_out;                  // (B,D)   fp32
    float* ws  = (float*)d_ws;

    float* part   = ws;                       // B*TC*D, reused as acc_part
    float* s_mean = ws + (size_t)BB * TC * DD;
    float* C1     = s_mean + (size_t)BB * DD;
    float* ml     = C1 + (size_t)BB * DD;

    mean_part_kernel <<<BB * TC, 128, 0, stream>>>(x, part);
    mean_final_kernel<<<BB,      128, 0, stream>>>(part, s_mean);
    small_gemm_kernel<<<1,       512, 0, stream>>>(s_mean, kern, C1);
    flash_kernel     <<<BB * NC, 128, 0, stream>>>(x, C1, part, ml);
    merge_kernel     <<<BB,      128, 0, stream>>>(part, ml, out);
}